// GaussianFlowOccRasterizer_60241211293723
// MI455X (gfx1250) — compile-verified
//
#include <hip/hip_runtime.h>

// Problem constants from the reference:
// values: (B=2, T=3, N=6, H=256, W=704, C=17) float32
// indices: (P, 3) int32   coors: (P, 2) float32   out: (P, 17) float32
#define BB 2
#define TT 3
#define NN 6
#define HH 256
#define WW 704
#define CC 17
#define PP 500000

__global__ __launch_bounds__(256) void bilinear_gather_kernel(
    const float* __restrict__ values,
    const int*   __restrict__ indices,
    const float* __restrict__ coors,
    float*       __restrict__ out)
{
    const int p = blockIdx.x * blockDim.x + threadIdx.x;
    if (p >= PP) return;

    // ---- per-point metadata: read-once streams -> non-temporal loads so the
    // 10 MB of indices/coors never displace `values` lines in the 192 MB L2.
    const int b = __builtin_nontemporal_load(indices + p * 3 + 0);
    const int t = __builtin_nontemporal_load(indices + p * 3 + 1);
    const int n = __builtin_nontemporal_load(indices + p * 3 + 2);
    const int flat = (b * TT + t) * NN + n;

    const float cy = __builtin_nontemporal_load(coors + p * 2 + 0);
    const float cx = __builtin_nontemporal_load(coors + p * 2 + 1);
    const float iy = cy - 0.5f;
    const float ix = cx - 0.5f;

    const float x0f = floorf(ix);
    const float y0f = floorf(iy);
    const float wx = ix - x0f;
    const float wy = iy - y0f;
    const int x0 = (int)x0f;
    const int y0 = (int)y0f;
    const int x1 = x0 + 1;
    const int y1 = y0 + 1;

    // Clipped coordinates -> always-safe addresses; OOB handled by zero weights.
    const int x0c = min(max(x0, 0), WW - 1);
    const int x1c = min(max(x1, 0), WW - 1);
    const int y0c = min(max(y0, 0), HH - 1);
    const int y1c = min(max(y1, 0), HH - 1);

    const bool bx0 = (x0 >= 0) & (x0 < WW);
    const bool bx1 = (x1 >= 0) & (x1 < WW);
    const bool by0 = (y0 >= 0) & (y0 < HH);
    const bool by1 = (y1 >= 0) & (y1 < HH);

    const float omwx = 1.0f - wx;
    const float omwy = 1.0f - wy;
    const float w00 = (bx0 & by0) ? omwx * omwy : 0.0f;
    const float w01 = (bx1 & by0) ? wx   * omwy : 0.0f;
    const float w10 = (bx0 & by1) ? omwx * wy   : 0.0f;
    const float w11 = (bx1 & by1) ? wx   * wy   : 0.0f;

    const float* __restrict__ img =
        values + (size_t)flat * ((size_t)HH * WW * CC);
    const float* __restrict__ p00 = img + ((size_t)y0c * WW + x0c) * CC;
    const float* __restrict__ p01 = img + ((size_t)y0c * WW + x1c) * CC;
    const float* __restrict__ p10 = img + ((size_t)y1c * WW + x0c) * CC;
    const float* __restrict__ p11 = img + ((size_t)y1c * WW + x1c) * CC;

    // gfx1250: global_prefetch_b8 — kick both scattered row segments toward
    // L2/WGP$ as soon as addresses exist (x0/x1 corners share a row, so one
    // prefetch per row covers most of its ~136 B span). `values` keeps RT
    // temporal policy: it is the only buffer with cross-point L2 reuse.
    __builtin_prefetch(p00, 0, 3);
    __builtin_prefetch(p10, 0, 3);

    // ---- 68 independent gather loads + 4-term blend per channel ----
    // No cross-iteration deps: backend clauses them into wide unaligned
    // global_load packets and retires with split s_wait_loadcnt.
    float acc[CC];
#pragma unroll
    for (int c = 0; c < CC; ++c) {
        const float v00 = p00[c];
        const float v01 = p01[c];
        const float v10 = p10[c];
        const float v11 = p11[c];
        acc[c] = fmaf(w00, v00,
                 fmaf(w01, v01,
                 fmaf(w10, v10,
                      w11 * v11)));
    }

    // Output is write-once (34 MB stream): non-temporal stores keep it from
    // evicting the gather working set out of L2.
    float* __restrict__ o = out + (size_t)p * CC;
#pragma unroll
    for (int c = 0; c < CC; ++c) {
        __builtin_nontemporal_store(acc[c], o + c);
    }
}

extern "C" void kernel_launch(void* const* d_in, const int* in_sizes, int n_in,
                              void* d_out, int out_size, void* d_ws, size_t ws_size,
                              hipStream_t stream) {
    (void)in_sizes; (void)n_in; (void)d_ws; (void)ws_size; (void)out_size;
    const float* values  = (const float*)d_in[0];
    const int*   indices = (const int*)  d_in[1];
    const float* coors   = (const float*)d_in[2];
    float*       out     = (float*)d_out;

    const int threads = 256;                    // 8 wave32 waves per block
    const int blocks  = (PP + threads - 1) / threads;
    bilinear_gather_kernel<<<blocks, threads, 0, stream>>>(values, indices, coors, out);
}